// FeatureGCN_43430709297954
// MI455X (gfx1250) — compile-verified
//
#include <hip/hip_runtime.h>
#include <hip/hip_bf16.h>

typedef __attribute__((ext_vector_type(2))) float v2f;
typedef __attribute__((ext_vector_type(8))) float v8f;

// ---------------------------------------------------------------------------
// Degree kernels: deg = 1 (self loop) + count(dst); dinv = rsqrt(deg)
// ---------------------------------------------------------------------------
__global__ void gcn_deg_init(float* __restrict__ deg, int n) {
    int i = blockIdx.x * blockDim.x + threadIdx.x;
    if (i < n) deg[i] = 1.0f;
}

__global__ void gcn_deg_count(const long long* __restrict__ dst, float* __restrict__ deg,
                              long long E) {
    long long i = (long long)blockIdx.x * blockDim.x + threadIdx.x;
    if (i < E) atomicAdd(&deg[dst[i]], 1.0f);
}

__global__ void gcn_deg_rsqrt(float* __restrict__ deg, int n) {
    int i = blockIdx.x * blockDim.x + threadIdx.x;
    if (i < n) deg[i] = rsqrtf(deg[i]);   // deg >= 1 always (self loops)
}

// ---------------------------------------------------------------------------
// WMMA fp32 GEMM: C[M,Nc] = A[M,K] * B[K,Nc].
// One 16x16 output tile per wave, V_WMMA_F32_16X16X4_F32, K stepped by 4.
// blockDim.x = (Nc/16)*32; gridDim.x = M/16.  M,K,Nc multiples of 16 (K of 4).
// ---------------------------------------------------------------------------
__global__ void __launch_bounds__(256)
gcn_gemm_wmma_f32(const float* __restrict__ A, const float* __restrict__ B,
                  float* __restrict__ C, int K, int Nc) {
    const int lane = threadIdx.x & 31;
    const int wave = threadIdx.x >> 5;
    const int m0   = blockIdx.x << 4;
    const int n0   = wave << 4;
    if (n0 >= Nc) return;                 // wave-uniform; EXEC stays all-ones otherwise

    const int half = lane >> 4;           // 0: K=0,1  1: K=2,3   (ISA A 16x4 layout)
    const int r    = lane & 15;

    const float* __restrict__ arow = A + (size_t)(m0 + r) * (size_t)K;
    v8f acc = {};

    for (int k = 0; k < K; k += 4) {
        v2f a, b;
        // A fragment: lane<16 -> {K=k, K=k+1}; lane>=16 -> {K=k+2, K=k+3}, row m0+r
        a.x = arow[k + 2 * half + 0];
        a.y = arow[k + 2 * half + 1];
        // B fragment: VGPR0: rows k (lanes 0-15) / k+2 (lanes 16-31); VGPR1: k+1 / k+3
        b.x = B[(size_t)(k + 2 * half + 0) * (size_t)Nc + (n0 + r)];
        b.y = B[(size_t)(k + 2 * half + 1) * (size_t)Nc + (n0 + r)];
        acc = __builtin_amdgcn_wmma_f32_16x16x4_f32(
            /*neg_a=*/false, a, /*neg_b=*/false, b,
            /*c_mod=*/(short)0, acc, /*reuse_a=*/false, /*reuse_b=*/false);
    }

    // D layout: VGPR v -> row m0+v (lanes 0-15) / m0+v+8 (lanes 16-31), col n0+lane%16
#pragma unroll
    for (int v = 0; v < 8; ++v) {
        int row = m0 + v + 8 * half;
        C[(size_t)row * (size_t)Nc + (n0 + r)] = acc[v];
    }
}

// ---------------------------------------------------------------------------
// Self-loop init: out[i,f] = dinv[i]^2 * h[i,f]   (fully overwrites poison)
// ---------------------------------------------------------------------------
template <int F>
__global__ void gcn_agg_init(const float* __restrict__ h, const float* __restrict__ dinv,
                             float* __restrict__ out, int n) {
    long long idx = (long long)blockIdx.x * blockDim.x + threadIdx.x;
    if (idx < (long long)n * F) {
        int node = (int)(idx / F);
        float di  = dinv[node];
        out[idx] = di * di * h[idx];
    }
}

// ---------------------------------------------------------------------------
// Edge scatter: one edge per wave32, lanes stride the feature dim.
// out[dst] += dinv[src]*dinv[dst] * h[src].  h and out are L2-resident
// (51.2 MB / 25.6 MB each vs 192 MB L2), so the random access stays on-chip.
// ---------------------------------------------------------------------------
template <int F>
__global__ void __launch_bounds__(256)
gcn_agg_edges(const float* __restrict__ h, const long long* __restrict__ src,
              const long long* __restrict__ dst, const float* __restrict__ dinv,
              float* __restrict__ out, long long E) {
    const int  lane = threadIdx.x & 31;
    long long  e    = (long long)blockIdx.x * (blockDim.x >> 5) + (threadIdx.x >> 5);
    if (e >= E) return;

    long long s = src[e];
    long long d = dst[e];
    float w = dinv[s] * dinv[d];
    const float* __restrict__ hs = h + s * F;
    float* __restrict__ od = out + d * F;
#pragma unroll
    for (int f = lane; f < F; f += 32) {
        atomicAdd(&od[f], w * hs[f]);
    }
}

// ---------------------------------------------------------------------------
// Bias (+ optional ReLU), in place.
// ---------------------------------------------------------------------------
template <int F, bool RELU>
__global__ void gcn_bias_act(float* __restrict__ x, const float* __restrict__ b, int n) {
    long long idx = (long long)blockIdx.x * blockDim.x + threadIdx.x;
    if (idx < (long long)n * F) {
        int f = (int)(idx % F);
        float v = x[idx] + b[f];
        if (RELU) v = fmaxf(v, 0.0f);
        x[idx] = v;
    }
}

// ---------------------------------------------------------------------------
// Launch
// ---------------------------------------------------------------------------
extern "C" void kernel_launch(void* const* d_in, const int* in_sizes, int n_in,
                              void* d_out, int out_size, void* d_ws, size_t ws_size,
                              hipStream_t stream) {
    const float*     x   = (const float*)d_in[0];
    const long long* ei  = (const long long*)d_in[1];
    const float*     W1  = (const float*)d_in[2];
    const float*     b1  = (const float*)d_in[3];
    const float*     W2  = (const float*)d_in[4];
    const float*     b2  = (const float*)d_in[5];

    const int       N = in_sizes[0] / 256;     // 100000 (multiple of 16)
    const long long E = in_sizes[1] / 2;       // 1600000
    const long long* src = ei;
    const long long* dst = ei + E;

    // Workspace layout: dinv[N] | h1[N*128] | agg1[N*128] | h2[N*64]
    float* dinv = (float*)d_ws;
    float* h1   = dinv + N;
    float* agg1 = h1 + (size_t)N * 128;
    float* h2   = agg1 + (size_t)N * 128;
    float* out  = (float*)d_out;

    // --- degrees on A+I ---
    gcn_deg_init<<<(N + 255) / 256, 256, 0, stream>>>(dinv, N);
    gcn_deg_count<<<(int)((E + 255) / 256), 256, 0, stream>>>(dst, dinv, E);
    gcn_deg_rsqrt<<<(N + 255) / 256, 256, 0, stream>>>(dinv, N);

    // --- layer 1: h1 = x @ W1 (K=256, Nc=128) ---
    gcn_gemm_wmma_f32<<<N / 16, (128 / 16) * 32, 0, stream>>>(x, W1, h1, 256, 128);
    {
        long long tot = (long long)N * 128;
        gcn_agg_init<128><<<(int)((tot + 255) / 256), 256, 0, stream>>>(h1, dinv, agg1, N);
        gcn_agg_edges<128><<<(int)((E + 7) / 8), 256, 0, stream>>>(h1, src, dst, dinv, agg1, E);
        gcn_bias_act<128, true><<<(int)((tot + 255) / 256), 256, 0, stream>>>(agg1, b1, N);
    }

    // --- layer 2: h2 = agg1 @ W2 (K=128, Nc=64) ---
    gcn_gemm_wmma_f32<<<N / 16, (64 / 16) * 32, 0, stream>>>(agg1, W2, h2, 128, 64);
    {
        long long tot = (long long)N * 64;
        gcn_agg_init<64><<<(int)((tot + 255) / 256), 256, 0, stream>>>(h2, dinv, out, N);
        gcn_agg_edges<64><<<(int)((E + 7) / 8), 256, 0, stream>>>(h2, src, dst, dinv, out, E);
        gcn_bias_act<64, false><<<(int)((tot + 255) / 256), 256, 0, stream>>>(out, b2, N);
    }
}